// SSDChunkwiseScan_44856638439973
// MI455X (gfx1250) — compile-verified
//
#include <hip/hip_runtime.h>

// Problem constants (match reference)
#define BB  2
#define LL  4096
#define HH  16
#define DD  128
#define CC  64
#define NCH (LL / CC)              // 64 chunks

// Output layout in d_out (floats): Y then chunk_states
#define Y_ELEMS   ((size_t)BB * LL * HH * DD * 2)          // 33,554,432
// Workspace layout (floats)
#define WS_TOTALA 0                                        // (b,n,h,2) complex totalA
#define WS_FINALH ((size_t)BB * NCH * HH * 2)              // then finalH (b,n,h,d,2)

// ---- Pass-1 LDS layout (floats) -------------------------------------------
#define S1_K     0            // [64][256] K tile                 64 KB
#define S1_V     16384        // [64][128] V tile                 32 KB
#define S1_COEF  24576        // [4][64]  a11,a12,beta,beta*dt     1 KB
#define S1_FLOATS 24832       // 99,328 B  -> 3 blocks / 320KB WGP

// ---- Pass-3 LDS layout (floats) -------------------------------------------
#define S3_K     0            // [64][256] K tile                 64 KB
#define S3_CORR  16384        // [64][128][2] WMMA correction     64 KB
#define S3_COEF  32768        // [4][64]
#define S3_CX    33024        // [64] cumulative A (complex re)
#define S3_CY    33088        // [64] cumulative A (complex im)
#define S3_SS    33152        // [128][2] entering state          1 KB
#define S3_FLOATS 33408       // 133,632 B -> 2 blocks / WGP

typedef __attribute__((ext_vector_type(2))) float v2f;
typedef __attribute__((ext_vector_type(8))) float v8f;

// ---- CDNA5 async global->LDS (ASYNCcnt-tracked data mover path) ------------
__device__ __forceinline__ unsigned ldsOffset(void* p) {
  return (unsigned)(unsigned long long)(uintptr_t)
         ((__attribute__((address_space(3))) void*)p);
}

__device__ __forceinline__ void asyncLoadB128(unsigned lds_byte_off, const void* gaddr) {
  asm volatile("global_load_async_to_lds_b128 %0, %1, off"
               :: "v"(lds_byte_off), "v"((unsigned long long)(uintptr_t)gaddr)
               : "memory");
}

__device__ __forceinline__ void waitAsync() {
  asm volatile("s_wait_asynccnt 0x0" ::: "memory");
}

// Stage K tile (64 rows x 1KB): one B128 per lane, 2 rows/issue, 32 issues.
__device__ __forceinline__ void stageK(const float* Kbase, float* sK, int tid) {
  const int sub = tid >> 6, q = tid & 63;
  #pragma unroll
  for (int tt = 0; tt < 32; ++tt) {
    const int t = tt * 2 + sub;
    asyncLoadB128(ldsOffset(sK + t * 256 + q * 4),
                  Kbase + (size_t)t * (HH * DD * 2) + q * 4);
  }
}
// Stage V tile (64 rows x 512B): 4 rows/issue, 16 issues.
__device__ __forceinline__ void stageV(const float* Vbase, float* sV, int tid) {
  const int sub = tid >> 5, q = tid & 31;
  #pragma unroll
  for (int tt = 0; tt < 16; ++tt) {
    const int t = tt * 4 + sub;
    asyncLoadB128(ldsOffset(sV + t * 128 + q * 4),
                  Vbase + (size_t)t * (HH * DD) + q * 4);
  }
}

// Per-t scalar coefficients -> coef[0..63]=a11 [64..]=a12 [128..]=beta [192..]=beta*dt
__device__ __forceinline__ void buildCoeffs(const float* alpha, const float* omega,
                                            const float* dtp, const float* beta,
                                            size_t rowScalar, float* coef, int tid) {
  if (tid < CC) {
    const size_t gi = rowScalar + (size_t)tid * HH;
    const float al = alpha[gi], om = omega[gi], dtv = dtp[gi], be = beta[gi];
    const float tau = 0.5f * dtv;
    const float ta = tau * al, tw = tau * om;
    const float inv = 1.0f / ((1.0f + ta) * (1.0f + ta) + tw * tw + 1e-6f);
    coef[tid]       = ((1.0f + ta) * (1.0f - ta) - tw * tw) * inv;
    coef[64 + tid]  = 2.0f * tw * inv;
    coef[128 + tid] = be;
    coef[192 + tid] = be * dtv;
  }
}

// ---------------------------------------------------------------------------
// Pass 1: per-chunk summaries (total_A as complex, final_h)
// ---------------------------------------------------------------------------
__global__ __launch_bounds__(DD)
void k_chunk_summary(const float* __restrict__ alpha, const float* __restrict__ omega,
                     const float* __restrict__ dtp,   const float* __restrict__ K,
                     const float* __restrict__ V,     const float* __restrict__ beta,
                     float* __restrict__ ws) {
  extern __shared__ float smem[];
  float* sK = smem + S1_K;
  float* sV = smem + S1_V;
  float* coef = smem + S1_COEF;

  const int tid = threadIdx.x;
  const int blk = blockIdx.x;
  const int h = blk % HH;
  const int n = (blk / HH) % NCH;
  const int b = blk / (HH * NCH);

  const size_t rowScalar = (size_t)(b * LL + n * CC) * HH + h;
  const float* Kbase = K + rowScalar * (DD * 2);
  const float* Vbase = V + rowScalar * DD;

  buildCoeffs(alpha, omega, dtp, beta, rowScalar, coef, tid);
  stageK(Kbase, sK, tid);
  stageV(Vbase, sV, tid);
  waitAsync();
  __syncthreads();

  const int d = tid;
  float s0 = 0.f, s1 = 0.f;
  #pragma unroll 4
  for (int t = 0; t < CC; ++t) {
    const float2 k2 = ((const float2*)(sK + t * 256))[d];
    const float  v  = sV[t * 128 + d];
    const float a = coef[t], c = coef[64 + t];
    const float be = coef[128 + t], bd = coef[192 + t];
    const float kth = k2.x * s0 + k2.y * s1;
    const float bm  = be * kth;
    const float m0 = s0 - bm * k2.x;
    const float m1 = s1 - bm * k2.y;
    const float e0 =  a * m0 + c * m1;      // A_t = [[a,c],[-c,a]]
    const float e1 =  a * m1 - c * m0;
    const float inj = bd * v;
    s0 = e0 + inj * k2.x;
    s1 = e1 + inj * k2.y;
  }

  const size_t bnh = (size_t)(b * NCH + n) * HH + h;
  *(float2*)(ws + WS_FINALH + (bnh * DD + d) * 2) = make_float2(s0, s1);

  if (tid == 0) {
    float x = 1.f, y = 0.f;
    for (int t = 0; t < CC; ++t) {
      const float a = coef[t], c = coef[64 + t];
      const float nx = a * x - c * y;
      const float ny = a * y + c * x;
      x = nx; y = ny;
    }
    ws[WS_TOTALA + bnh * 2 + 0] = x;
    ws[WS_TOTALA + bnh * 2 + 1] = y;
  }
}

// ---------------------------------------------------------------------------
// Pass 2: sequential inter-chunk scan; writes entering states (output #2)
// ---------------------------------------------------------------------------
__global__ __launch_bounds__(DD)
void k_inter(const float* __restrict__ ws, float* __restrict__ states_out) {
  const int d = threadIdx.x;
  const int h = blockIdx.x % HH;
  const int b = blockIdx.x / HH;
  float s0 = 0.f, s1 = 0.f;
  for (int n = 0; n < NCH; ++n) {
    const size_t bnh = (size_t)(b * NCH + n) * HH + h;
    *(float2*)(states_out + (bnh * DD + d) * 2) = make_float2(s0, s1);
    const float tx = ws[WS_TOTALA + bnh * 2 + 0];
    const float ty = ws[WS_TOTALA + bnh * 2 + 1];
    const float2 f = *(const float2*)(ws + WS_FINALH + (bnh * DD + d) * 2);
    const float n0 =  tx * s0 + ty * s1 + f.x;
    const float n1 = -ty * s0 + tx * s1 + f.y;
    s0 = n0; s1 = n1;
  }
}

// ---------------------------------------------------------------------------
// Pass 3: rerun scan; correction Corr(128x128) = A(128x2) @ S(2x128) via
// v_wmma_f32_16x16x4_f32 (K padded 2->4), staged through LDS; stream Y out.
// ---------------------------------------------------------------------------
__global__ __launch_bounds__(DD)
void k_output(const float* __restrict__ alpha, const float* __restrict__ omega,
              const float* __restrict__ dtp,   const float* __restrict__ K,
              const float* __restrict__ V,     const float* __restrict__ beta,
              const float* __restrict__ states_in, float* __restrict__ Y) {
  extern __shared__ float smem[];
  float* sK    = smem + S3_K;
  float* sCorr = smem + S3_CORR;
  float* coef  = smem + S3_COEF;
  float* sCX   = smem + S3_CX;
  float* sCY   = smem + S3_CY;
  float* sS    = smem + S3_SS;

  const int tid = threadIdx.x;
  const int blk = blockIdx.x;
  const int h = blk % HH;
  const int n = (blk / HH) % NCH;
  const int b = blk / (HH * NCH);

  const size_t rowScalar = (size_t)(b * LL + n * CC) * HH + h;
  const float* Kbase = K + rowScalar * (DD * 2);
  const float* Vbase = V + rowScalar * DD;

  buildCoeffs(alpha, omega, dtp, beta, rowScalar, coef, tid);
  stageK(Kbase, sK, tid);

  // entering state for this d -> LDS (for WMMA B-operand build)
  const size_t bnh = (size_t)(b * NCH + n) * HH + h;
  const float2 si = ((const float2*)(states_in + bnh * DD * 2))[tid];
  *(float2*)(sS + tid * 2) = si;

  waitAsync();
  __syncthreads();

  if (tid == 0) {
    float x = 1.f, y = 0.f;
    for (int t = 0; t < CC; ++t) {
      const float a = coef[t], c = coef[64 + t];
      const float nx = a * x - c * y;
      const float ny = a * y + c * x;
      x = nx; y = ny;
      sCX[t] = x; sCY[t] = y;
    }
  }
  __syncthreads();

  // ---- WMMA correction: rows r = i*64+t, cols = d ----
  // A row r: i==0 -> [cx[t], cy[t]] ; i==1 -> [-cy[t], cx[t]] ; K2,K3 = 0
  {
    const int wave = tid >> 5;
    const int lane = tid & 31;
    const int lm   = lane & 15;
    const bool lo  = lane < 16;           // lanes 16..31 carry K=2,3 (zero pad)
    #pragma unroll
    for (int mt2 = 0; mt2 < 2; ++mt2) {
      const int mt = wave * 2 + mt2;      // M-tile 0..7
      v2f a = {0.f, 0.f};
      if (lo) {
        const int r = mt * 16 + lm;
        const int t = r & 63;
        const float cx = sCX[t], cy = sCY[t];
        const bool i1 = (r >> 6) != 0;
        a.x = i1 ? -cy : cx;              // K=0
        a.y = i1 ?  cx : cy;              // K=1
      }
      #pragma unroll
      for (int nt = 0; nt < 8; ++nt) {    // N-tile 0..7
        v2f bv = {0.f, 0.f};
        if (lo) {
          const int d = nt * 16 + lm;
          bv.x = sS[d * 2 + 0];           // K=0 row: s0
          bv.y = sS[d * 2 + 1];           // K=1 row: s1
        }
        v8f cacc = {};
        cacc = __builtin_amdgcn_wmma_f32_16x16x4_f32(
            false, a, false, bv, (short)0, cacc, false, false);
        // D layout: VGPR j -> row M = j + (lane>=16 ? 8 : 0), col N = lane%16
        const int mof = lo ? 0 : 8;
        #pragma unroll
        for (int j = 0; j < 8; ++j) {
          const int r = mt * 16 + j + mof;
          const int t = r & 63;
          const int i = r >> 6;
          const int d = nt * 16 + lm;
          sCorr[(t * DD + d) * 2 + i] = cacc[j];
        }
      }
    }
  }
  __syncthreads();

  // ---- scan + Y writeout (V read straight from global, coalesced) ----
  const int d = tid;
  float s0 = 0.f, s1 = 0.f;
  float* Ybase = Y + rowScalar * (DD * 2);
  #pragma unroll 4
  for (int t = 0; t < CC; ++t) {
    const float2 k2 = ((const float2*)(sK + t * 256))[d];
    const float  v  = Vbase[(size_t)t * (HH * DD) + d];
    const float a = coef[t], c = coef[64 + t];
    const float be = coef[128 + t], bd = coef[192 + t];
    const float kth = k2.x * s0 + k2.y * s1;
    const float bm  = be * kth;
    const float m0 = s0 - bm * k2.x;
    const float m1 = s1 - bm * k2.y;
    const float e0 =  a * m0 + c * m1;
    const float e1 =  a * m1 - c * m0;
    const float inj = bd * v;
    s0 = e0 + inj * k2.x;
    s1 = e1 + inj * k2.y;
    const float2 corr = ((const float2*)sCorr)[t * DD + d];
    ((float2*)(Ybase + (size_t)t * (HH * DD * 2)))[d] =
        make_float2(s0 + corr.x, s1 + corr.y);
  }
}

// ---------------------------------------------------------------------------
extern "C" void kernel_launch(void* const* d_in, const int* in_sizes, int n_in,
                              void* d_out, int out_size, void* d_ws, size_t ws_size,
                              hipStream_t stream) {
  const float* alpha = (const float*)d_in[0];
  const float* omega = (const float*)d_in[1];
  const float* dtp   = (const float*)d_in[2];
  const float* K     = (const float*)d_in[3];
  const float* V     = (const float*)d_in[4];
  const float* beta  = (const float*)d_in[5];

  float* Y      = (float*)d_out;
  float* states = Y + Y_ELEMS;
  float* ws     = (float*)d_ws;

  k_chunk_summary<<<BB * NCH * HH, DD, S1_FLOATS * sizeof(float), stream>>>(
      alpha, omega, dtp, K, V, beta, ws);
  k_inter<<<BB * HH, DD, 0, stream>>>(ws, states);
  k_output<<<BB * NCH * HH, DD, S3_FLOATS * sizeof(float), stream>>>(
      alpha, omega, dtp, K, V, beta, states, Y);
}